// GCN_9698036155051
// MI455X (gfx1250) — compile-verified
//
#include <hip/hip_runtime.h>
#include <hip/hip_bf16.h>
#include <math.h>

#define N_NODES 50000
#define N_EDGES 800000
#define F_IN    512
#define HID     256
#define N_CLS   64

#define K_MAX   512
#define A_PAD   4              // floats; keeps 16B rows for async B128, spreads LDS banks

typedef __attribute__((ext_vector_type(2))) float v2f;
typedef __attribute__((ext_vector_type(8))) float v8f;
typedef __attribute__((address_space(3))) float lds_float;

// ---------------- degree / normalization ----------------

__global__ __launch_bounds__(256) void k_zero_u32(unsigned* __restrict__ p, int n) {
    int i = blockIdx.x * blockDim.x + threadIdx.x;
    if (i < n) p[i] = 0u;
}

__global__ __launch_bounds__(256) void k_deg(const int* __restrict__ dst,
                                             unsigned* __restrict__ deg, int nE) {
    int e = blockIdx.x * blockDim.x + threadIdx.x;
    if (e < nE) atomicAdd(&deg[dst[e]], 1u);
}

__global__ __launch_bounds__(256) void k_dinv(const unsigned* __restrict__ deg,
                                              float* __restrict__ dinv, int n) {
    int i = blockIdx.x * blockDim.x + threadIdx.x;
    if (i < n) {
        // +1 for the self loop -> deg >= 1 always
        dinv[i] = rsqrtf((float)(deg[i] + 1u));
    }
}

// ---------------- fp32 WMMA GEMM: C[M,N] = A[M,K] * B[K,N] ----------------
// One block owns one 16-row stripe of A, staged once into LDS with async
// global->LDS B128 copies (ASYNCcnt). Each wave sweeps column tiles with
// V_WMMA_F32_16X16X4_F32, reading A fragments from LDS (ds_load_b64,
// bank-conflict-free via A_PAD) and B columns from L2.
// Requires: M,N multiples of 16, K multiple of 4 and <= K_MAX,
// gridDim.x == M/16, full wave32 blocks (EXEC all-ones).

__global__ __launch_bounds__(256) void k_gemm_f32_wmma(
    const float* __restrict__ A, const float* __restrict__ B, float* __restrict__ C,
    int K, int N) {
    __shared__ float As[16 * (K_MAX + A_PAD)];

    const int tm     = blockIdx.x;          // row tile
    const int lane   = threadIdx.x & 31;
    const int wave   = threadIdx.x >> 5;
    const int nWaves = blockDim.x >> 5;
    const int tilesN = N >> 4;
    const int ldA    = K + A_PAD;

    // ---- stage A tile (16 x K) into LDS via async copies ----
    {
        const int kq = K >> 2;              // float4s per row
        const int nVec = 16 * kq;
        for (int i = threadIdx.x; i < nVec; i += blockDim.x) {
            int row = i / kq;
            int c4  = i - row * kq;
            const float* g = A + (size_t)(tm * 16 + row) * K + c4 * 4;
            lds_float* l = (lds_float*)&As[row * ldA + c4 * 4];
            asm volatile("global_load_async_to_lds_b128 %0, %1, off"
                         :: "v"(l), "v"(g) : "memory");
        }
        asm volatile("s_wait_asynccnt 0x0" ::: "memory");
        __syncthreads();
    }

    const int half = lane >> 4;             // 0 = lanes 0-15, 1 = lanes 16-31
    const int l15  = lane & 15;
    const float* arow = &As[l15 * ldA + 2 * half];

    for (int tn = wave; tn < tilesN; tn += nWaves) {
        // B fragment (4x16 f32): lane N = l15; VGPR0/1 hold K rows 2*half, 2*half+1
        const float* bcol = B + (size_t)(2 * half) * N + tn * 16 + l15;
        v8f acc = {};
        for (int k0 = 0; k0 < K; k0 += 4) {
            v2f a, b;
            a.x = arow[k0];                 // LDS: K = k0 + 2*half
            a.y = arow[k0 + 1];
            b.x = bcol[(size_t)k0 * N];
            b.y = bcol[(size_t)(k0 + 1) * N];
            acc = __builtin_amdgcn_wmma_f32_16x16x4_f32(
                /*neg_a=*/false, a, /*neg_b=*/false, b,
                /*c_mod=*/(short)0, acc, /*reuse_a=*/false, /*reuse_b=*/false);
        }
        // C/D 16x16 f32: VGPR r holds M = r + 8*half, N = l15
        float* crow = C + (size_t)(tm * 16 + 8 * half) * N + tn * 16 + l15;
#pragma unroll
        for (int r = 0; r < 8; ++r) crow[(size_t)r * N] = acc[r];
    }
}

// ---------------- aggregation ----------------

// Seed accumulator with the self-loop term: S[i,:] = dinv[i]^2 * T[i,:]
__global__ __launch_bounds__(256) void k_selfloop(const float* __restrict__ T,
                                                  float* __restrict__ S,
                                                  const float* __restrict__ dinv, int F) {
    int i = blockIdx.x;
    float w = dinv[i] * dinv[i];
    const float* t = T + (size_t)i * F;
    float* s = S + (size_t)i * F;
    for (int f = threadIdx.x; f < F; f += blockDim.x) s[f] = w * t[f];
}

// One wave per edge: S[dst,:] += dinv[src]*dinv[dst] * T[src,:]
// float4 gathers (global_load_b128) + per-float global_atomic_add_f32.
__global__ __launch_bounds__(256) void k_scatter(const float* __restrict__ T,
                                                 float* __restrict__ S,
                                                 const int* __restrict__ src,
                                                 const int* __restrict__ dst,
                                                 const float* __restrict__ dinv,
                                                 int nE, int F) {
    const int lane = threadIdx.x & 31;
    const int wave = threadIdx.x >> 5;
    int e = blockIdx.x * (blockDim.x >> 5) + wave;
    if (e >= nE) return;
    int s = src[e], d = dst[e];
    float w = dinv[s] * dinv[d];
    const float4* t4 = (const float4*)(T + (size_t)s * F);
    float* srow = S + (size_t)d * F;
    const int F4 = F >> 2;
    for (int q = lane; q < F4; q += 32) {
        float4 v = t4[q];
        int f = q * 4;
        atomicAdd(&srow[f + 0], w * v.x);
        atomicAdd(&srow[f + 1], w * v.y);
        atomicAdd(&srow[f + 2], w * v.z);
        atomicAdd(&srow[f + 3], w * v.w);
    }
}

// In-place H = relu(S + b)
__global__ __launch_bounds__(256) void k_bias_relu(float* __restrict__ S,
                                                   const float* __restrict__ b, int F) {
    int i = blockIdx.x;
    float* s = S + (size_t)i * F;
    for (int f = threadIdx.x; f < F; f += blockDim.x) {
        float v = s[f] + b[f];
        s[f] = v > 0.0f ? v : 0.0f;
    }
}

// ---------------- log-softmax over 64 classes (one wave per row) ----------------

__global__ __launch_bounds__(256) void k_logsoftmax(const float* __restrict__ S,
                                                    const float* __restrict__ b,
                                                    float* __restrict__ out, int nRows) {
    const int lane = threadIdx.x & 31;
    const int wave = threadIdx.x >> 5;
    int row = blockIdx.x * (blockDim.x >> 5) + wave;
    if (row >= nRows) return;
    const float* s = S + (size_t)row * N_CLS;
    float v0 = s[lane]      + b[lane];
    float v1 = s[lane + 32] + b[lane + 32];
    float m = fmaxf(v0, v1);
#pragma unroll
    for (int off = 16; off > 0; off >>= 1)
        m = fmaxf(m, __shfl_xor(m, off, 32));
    float sum = expf(v0 - m) + expf(v1 - m);
#pragma unroll
    for (int off = 16; off > 0; off >>= 1)
        sum += __shfl_xor(sum, off, 32);
    float lse = m + logf(sum);
    float* o = out + (size_t)row * N_CLS;
    o[lane]      = v0 - lse;
    o[lane + 32] = v1 - lse;
}

// ---------------- driver ----------------

extern "C" void kernel_launch(void* const* d_in, const int* in_sizes, int n_in,
                              void* d_out, int out_size, void* d_ws, size_t ws_size,
                              hipStream_t stream) {
    const float* x  = (const float*)d_in[0];
    const int*   ei = (const int*)d_in[1];      // [2, N_EDGES] row-major
    const float* W1 = (const float*)d_in[2];
    const float* b1 = (const float*)d_in[3];
    const float* W2 = (const float*)d_in[4];
    const float* b2 = (const float*)d_in[5];
    const int* src = ei;
    const int* dst = ei + N_EDGES;
    float* out = (float*)d_out;

    // workspace layout (aliases: H overwrites S1; T2/S2 reuse dead T1 buffer)
    char* w = (char*)d_ws;
    auto align256 = [](size_t v) { return (v + 255) & ~(size_t)255; };
    unsigned* deg  = (unsigned*)w;  w += align256((size_t)N_NODES * 4);
    float*    dinv = (float*)w;     w += align256((size_t)N_NODES * 4);
    float*    T1   = (float*)w;     w += (size_t)N_NODES * HID * 4;
    float*    S1   = (float*)w;     w += (size_t)N_NODES * HID * 4;
    float*    T2   = T1;                                   // T1 dead after scatter1
    float*    S2   = T1 + (size_t)N_NODES * N_CLS;         // fits in T1 buffer

    // normalization
    k_zero_u32<<<(N_NODES + 255) / 256, 256, 0, stream>>>(deg, N_NODES);
    k_deg<<<(N_EDGES + 255) / 256, 256, 0, stream>>>(dst, deg, N_EDGES);
    k_dinv<<<(N_NODES + 255) / 256, 256, 0, stream>>>(deg, dinv, N_NODES);

    // layer 1: T1 = X * W1 ; S1 = aggregate ; H = relu(S1 + b1) in place
    k_gemm_f32_wmma<<<N_NODES / 16, 256, 0, stream>>>(x, W1, T1, F_IN, HID);
    k_selfloop<<<N_NODES, 256, 0, stream>>>(T1, S1, dinv, HID);
    k_scatter<<<(N_EDGES + 7) / 8, 256, 0, stream>>>(T1, S1, src, dst, dinv, N_EDGES, HID);
    k_bias_relu<<<N_NODES, 256, 0, stream>>>(S1, b1, HID);

    // layer 2: T2 = H * W2 ; S2 = aggregate ; out = log_softmax(S2 + b2)
    k_gemm_f32_wmma<<<N_NODES / 16, 128, 0, stream>>>(S1, W2, T2, HID, N_CLS);
    k_selfloop<<<N_NODES, 256, 0, stream>>>(T2, S2, dinv, N_CLS);
    k_scatter<<<(N_EDGES + 7) / 8, 256, 0, stream>>>(T2, S2, src, dst, dinv, N_EDGES, N_CLS);
    k_logsoftmax<<<(N_NODES + 7) / 8, 256, 0, stream>>>(S2, b2, out, N_NODES);
}